// Attn_5978594476256
// MI455X (gfx1250) — compile-verified
//
#include <hip/hip_runtime.h>
#include <math.h>

typedef __attribute__((ext_vector_type(2))) float v2f;
typedef __attribute__((ext_vector_type(8))) float v8f;

#define S_LEN 2048
#define B_N   32
#define H_DIM 1024

// ---------------------------------------------------------------------------
// Kernel 1: v[b,n] = sum_k hidden[b,k] * W[k,n]   (A=hidden 32x1024, B=W 1024x1024)
// fp32 WMMA 16x16x4. One wave computes a 32x16 tile of v (two accumulators).
// Grid: 16 blocks x 4 waves -> 64 N-tiles of 16 = 1024 columns.
// ---------------------------------------------------------------------------
__global__ __launch_bounds__(128) void proj_hidden_wmma(
    const float* __restrict__ hidden,   // [B_N, H_DIM]
    const float* __restrict__ W,        // [H_DIM(out o), H_DIM(in h)] row-major
    float* __restrict__ v)              // [B_N, H_DIM]
{
    const int lane = threadIdx.x & 31;
    const int wave = threadIdx.x >> 5;
    const int n0   = blockIdx.x * 64 + wave * 16;   // N tile base
    const int half = lane >> 4;                     // 0: lanes 0-15, 1: lanes 16-31
    const int l    = lane & 15;

    v8f c0 = {};   // M = 0..15
    v8f c1 = {};   // M = 16..31

    for (int k = 0; k < H_DIM; k += 4) {
        // ISA 16x4 f32 A layout: reg0 = K=(k | k+2), reg1 = K=(k+1 | k+3)
        const int ka = k + half * 2;
        v2f a0, a1, bfr;
        a0.x = hidden[l * H_DIM + ka];
        a0.y = hidden[l * H_DIM + ka + 1];
        a1.x = hidden[(l + 16) * H_DIM + ka];
        a1.y = hidden[(l + 16) * H_DIM + ka + 1];
        // B 4x16 layout: reg0 = row (k | k+2), reg1 = row (k+1 | k+3), col = lane%16
        bfr.x = W[(size_t)(ka)     * H_DIM + n0 + l];
        bfr.y = W[(size_t)(ka + 1) * H_DIM + n0 + l];

        c0 = __builtin_amdgcn_wmma_f32_16x16x4_f32(false, a0, false, bfr,
                                                   (short)0, c0, false, false);
        c1 = __builtin_amdgcn_wmma_f32_16x16x4_f32(false, a1, false, bfr,
                                                   (short)0, c1, false, false);
    }

    // D layout: VGPR r -> lanes 0-15: M=r, lanes 16-31: M=r+8; N = lane%16
    #pragma unroll
    for (int r = 0; r < 8; ++r) {
        const int row = r + half * 8;
        v[(row)      * H_DIM + n0 + l] = c0[r];
        v[(row + 16) * H_DIM + n0 + l] = c1[r];
    }
}

// ---------------------------------------------------------------------------
// Kernel 2: energies[b,s] = enc[s,b,:] . v[b,:]
// 256-thread block = 8 waves; one batch b, 8 consecutive s per block.
// v[b,:] staged in LDS (4KB). Fully coalesced float4 streaming of enc (268MB
// total -> HBM-bound, ~11.5us at 23.3 TB/s).
// ---------------------------------------------------------------------------
__global__ __launch_bounds__(256) void energies_kernel(
    const float* __restrict__ enc,      // [S_LEN, B_N, H_DIM]
    const float* __restrict__ v,        // [B_N, H_DIM]
    float* __restrict__ e)              // [B_N, S_LEN]
{
    __shared__ float vsh[H_DIM];
    const int b     = blockIdx.x & (B_N - 1);
    const int stile = blockIdx.x >> 5;

    for (int i = threadIdx.x; i < H_DIM; i += 256)
        vsh[i] = v[b * H_DIM + i];
    __syncthreads();

    const int lane = threadIdx.x & 31;
    const int wave = threadIdx.x >> 5;
    const int s    = stile * 8 + wave;
    const float* row = enc + (size_t)s * (B_N * H_DIM) + (size_t)b * H_DIM;

    float acc = 0.0f;
    #pragma unroll
    for (int i = 0; i < 8; ++i) {
        const int h = i * 128 + lane * 4;           // 512B coalesced per wave-load
        float4 x = *(const float4*)(row + h);
        float4 w = *(const float4*)(vsh + h);
        acc = fmaf(x.x, w.x, acc);
        acc = fmaf(x.y, w.y, acc);
        acc = fmaf(x.z, w.z, acc);
        acc = fmaf(x.w, w.w, acc);
    }
    // wave32 cross-lane reduction
    #pragma unroll
    for (int off = 16; off > 0; off >>= 1)
        acc += __shfl_xor(acc, off, 32);

    if (lane == 0)
        e[b * S_LEN + s] = acc;
}

// ---------------------------------------------------------------------------
// Kernel 3: out[b,0,s] = log_softmax_s(e[b,s]). One block per batch.
// Note: the per-batch constant (attn_b . hidden[b]) cancels in log_softmax,
// so attn_b never enters the computation at all.
// ---------------------------------------------------------------------------
__global__ __launch_bounds__(256) void logsoftmax_kernel(
    const float* __restrict__ e,        // [B_N, S_LEN]
    float* __restrict__ out)            // [B_N, 1, S_LEN]
{
    __shared__ float row[S_LEN];
    __shared__ float redm[8];
    __shared__ float reds[8];

    const int b    = blockIdx.x;
    const int t    = threadIdx.x;
    const int lane = t & 31;
    const int wv   = t >> 5;

    float lmax = -INFINITY;
    for (int i = t; i < S_LEN; i += 256) {
        float x = e[b * S_LEN + i];
        row[i] = x;
        lmax = fmaxf(lmax, x);
    }
    #pragma unroll
    for (int off = 16; off > 0; off >>= 1)
        lmax = fmaxf(lmax, __shfl_xor(lmax, off, 32));
    if (lane == 0) redm[wv] = lmax;
    __syncthreads();

    float m = (t < 8) ? redm[t] : -INFINITY;
    if (wv == 0) {
        #pragma unroll
        for (int off = 4; off > 0; off >>= 1)
            m = fmaxf(m, __shfl_xor(m, off, 32));
        if (lane == 0) redm[0] = m;
    }
    __syncthreads();
    const float gmax = redm[0];

    float lsum = 0.0f;
    for (int i = t; i < S_LEN; i += 256)
        lsum += expf(row[i] - gmax);
    #pragma unroll
    for (int off = 16; off > 0; off >>= 1)
        lsum += __shfl_xor(lsum, off, 32);
    if (lane == 0) reds[wv] = lsum;
    __syncthreads();

    float s8 = (t < 8) ? reds[t] : 0.0f;
    if (wv == 0) {
        #pragma unroll
        for (int off = 4; off > 0; off >>= 1)
            s8 += __shfl_xor(s8, off, 32);
        if (lane == 0) reds[0] = s8;
    }
    __syncthreads();

    const float lse = gmax + logf(reds[0]);
    for (int i = t; i < S_LEN; i += 256)
        out[b * S_LEN + i] = row[i] - lse;
}

// ---------------------------------------------------------------------------
// d_in order: hidden [1,B,H], encoder_outputs [S,B,H], attn_W [H,H], attn_b [H]
// d_out: [B,1,S] float32. Workspace: v (128KB) + energies (256KB) = 384KB.
// ---------------------------------------------------------------------------
extern "C" void kernel_launch(void* const* d_in, const int* in_sizes, int n_in,
                              void* d_out, int out_size, void* d_ws, size_t ws_size,
                              hipStream_t stream) {
    const float* hidden = (const float*)d_in[0];
    const float* enc    = (const float*)d_in[1];
    const float* W      = (const float*)d_in[2];
    // d_in[3] (attn_b) provably cancels in log_softmax -> unused.
    (void)in_sizes; (void)n_in; (void)out_size; (void)ws_size;

    float* v = (float*)d_ws;
    float* e = (float*)((char*)d_ws + (size_t)B_N * H_DIM * sizeof(float));

    proj_hidden_wmma<<<16, 128, 0, stream>>>(hidden, W, v);
    energies_kernel<<<B_N * (S_LEN / 8), 256, 0, stream>>>(enc, v, e);
    logsoftmax_kernel<<<B_N, 256, 0, stream>>>(e, (float*)d_out);
}